// HamGraphConvolution_27745488732226
// MI455X (gfx1250) — compile-verified
//
#include <hip/hip_runtime.h>
#include <hip/hip_bf16.h>
#include <math.h>

typedef float v2f __attribute__((ext_vector_type(2)));
typedef float v8f __attribute__((ext_vector_type(8)));

#define ALPHAF 0.2f
#define STEPF  0.25f

__device__ __forceinline__ void atomicMaxF32(float* addr, float val) {
  // sign-split trick: int ordering matches float for >=0, reversed for <0
  if (val >= 0.0f) atomicMax((int*)addr, __float_as_int(val));
  else             atomicMin((unsigned int*)addr, __float_as_uint(val));
}

// ---------------------------------------------------------------------------
// Kernel 1: one wave per 16 rows.
//   p = x @ W^T via V_WMMA_F32_16X16X4_F32 (16 K-steps x 4 col-tiles),
//   then 4 Hamiltonian rotation steps elementwise, xt = q,
//   then per-head score dots s_src/s_dst from LDS.
// ---------------------------------------------------------------------------
__global__ __launch_bounds__(32)
void ham_wmma_kernel(const float* __restrict__ x, const float* __restrict__ W,
                     const float* __restrict__ avec,
                     float* __restrict__ xt,
                     float* __restrict__ s_src, float* __restrict__ s_dst)
{
  __shared__ float p_lds[16][64];
  __shared__ float q_lds[16][64];

  const int lane  = threadIdx.x;        // 0..31, wave32
  const int r0    = blockIdx.x * 16;    // N divisible by 16
  const int mh    = lane & 15;          // A row / B col within tile
  const int khalf = (lane >> 4) * 2;    // K sub-offset per half-wave (0 or 2)

  for (int jt = 0; jt < 4; ++jt) {
    v8f c = {};
    for (int kt = 0; kt < 16; ++kt) {
      const int kk = kt * 4 + khalf;
      v2f av, bv;
      // A (16x4 f32): lane<16 holds K=kk,kk+1 ; lane>=16 holds K=kk+2,kk+3
      av.x = x[(r0 + mh) * 64 + kk];
      av.y = x[(r0 + mh) * 64 + kk + 1];
      // B (4x16 f32): B[k][n] = W[n][k]  (computing x @ W^T)
      const int n = jt * 16 + mh;
      bv.x = W[n * 64 + kk];
      bv.y = W[n * 64 + kk + 1];
      c = __builtin_amdgcn_wmma_f32_16x16x4_f32(false, av, false, bv,
                                                (short)0, c, false, false);
    }
    // D layout: vgpr r, lanes 0-15 -> row r, lanes 16-31 -> row r+8; col = lane&15
    const int nD    = jt * 16 + mh;
    const int mbase = (lane >> 4) * 8;
#pragma unroll
    for (int r = 0; r < 8; ++r) p_lds[mbase + r][nD] = c[r];
  }
  __syncthreads();

  // Hamiltonian steps: q_{t+1} = q + s*p ; p_{t+1} = p - s*q  (both from old)
  for (int i = 0; i < 32; ++i) {
    const int idx = i * 32 + lane;        // 16*64 = 1024 elems / 32 lanes
    const int m = idx >> 6, d = idx & 63;
    float q = x[(r0 + m) * 64 + d];
    float p = p_lds[m][d];
#pragma unroll
    for (int s = 0; s < 4; ++s) {
      const float qn = q + STEPF * p;
      p = p - STEPF * q;
      q = qn;
    }
    q_lds[m][d] = q;
    xt[(size_t)(r0 + m) * 64 + d] = q;
  }
  __syncthreads();

  // s_src[n][h] = dot(xt[n, h*16:+16], a[:16]); s_dst uses a[16:32]
  for (int t = lane; t < 64; t += 32) {
    const int m = t >> 2, h = t & 3;
    float ss = 0.f, sd = 0.f;
#pragma unroll
    for (int j = 0; j < 16; ++j) {
      const float v = q_lds[m][h * 16 + j];
      ss += v * avec[j];
      sd += v * avec[16 + j];
    }
    s_src[(size_t)(r0 + m) * 4 + h] = ss;
    s_dst[(size_t)(r0 + m) * 4 + h] = sd;
  }
}

// ---------------------------------------------------------------------------
// Kernel 2: initialize seg_max = -inf, denom = 0, out = 0 (harness poisons)
// ---------------------------------------------------------------------------
__global__ void init_kernel(float* __restrict__ seg_max, float* __restrict__ denom,
                            float* __restrict__ out, int n4, int n64)
{
  const int i = blockIdx.x * blockDim.x + threadIdx.x;
  const int stride = gridDim.x * blockDim.x;
  for (int j = i; j < n4; j += stride) { seg_max[j] = -__builtin_inff(); denom[j] = 0.f; }
  for (int j = i; j < n64; j += stride) out[j] = 0.f;
}

// ---------------------------------------------------------------------------
// Kernel 3: per-edge leaky-relu score, atomic segment-max over col per head
// ---------------------------------------------------------------------------
__global__ __launch_bounds__(256)
void segmax_kernel(const int* __restrict__ ei, const float* __restrict__ s_src,
                   const float* __restrict__ s_dst, float* __restrict__ seg_max, int E)
{
  const int e = blockIdx.x * blockDim.x + threadIdx.x;
  if (e >= E) return;
  const int row = ei[e], col = ei[E + e];
  const float4 ss = *(const float4*)(s_src + (size_t)row * 4);
  const float4 sd = *(const float4*)(s_dst + (size_t)col * 4);
  float sc[4] = { ss.x + sd.x, ss.y + sd.y, ss.z + sd.z, ss.w + sd.w };
#pragma unroll
  for (int h = 0; h < 4; ++h) {
    const float v = sc[h] > 0.f ? sc[h] : ALPHAF * sc[h];
    atomicMaxF32(&seg_max[(size_t)col * 4 + h], v);
  }
}

// ---------------------------------------------------------------------------
// Kernel 4: e = exp(score - seg_max[col]); denom[col] += e
// ---------------------------------------------------------------------------
__global__ __launch_bounds__(256)
void denom_kernel(const int* __restrict__ ei, const float* __restrict__ s_src,
                  const float* __restrict__ s_dst, const float* __restrict__ seg_max,
                  float* __restrict__ denom, int E)
{
  const int e = blockIdx.x * blockDim.x + threadIdx.x;
  if (e >= E) return;
  const int row = ei[e], col = ei[E + e];
  const float4 ss = *(const float4*)(s_src + (size_t)row * 4);
  const float4 sd = *(const float4*)(s_dst + (size_t)col * 4);
  const float4 mx = *(const float4*)(seg_max + (size_t)col * 4);
  float sc[4] = { ss.x + sd.x, ss.y + sd.y, ss.z + sd.z, ss.w + sd.w };
  float mm[4] = { mx.x, mx.y, mx.z, mx.w };
#pragma unroll
  for (int h = 0; h < 4; ++h) {
    const float v = sc[h] > 0.f ? sc[h] : ALPHAF * sc[h];
    float m = mm[h];
    if (isinf(m)) m = 0.f;   // reference: where(isfinite(seg_max), seg_max, 0)
    atomicAdd(&denom[(size_t)col * 4 + h], __expf(v - m));
  }
}

// ---------------------------------------------------------------------------
// Kernel 5: scatter out[row] += w * xt[col]; 16 lanes per edge, float4 each.
// Per-edge softmax weight computed once per 16-lane group, shfl-broadcast.
// ---------------------------------------------------------------------------
__global__ __launch_bounds__(256)
void scatter_kernel(const int* __restrict__ ei, const float* __restrict__ s_src,
                    const float* __restrict__ s_dst, const float* __restrict__ seg_max,
                    const float* __restrict__ denom, const float* __restrict__ xt,
                    float* __restrict__ out, int E)
{
  const int gid = blockIdx.x * 256 + threadIdx.x;
  const int e   = gid >> 4;
  const int sub = threadIdx.x & 15;
  if (e >= E) return;
  const int row = ei[e], col = ei[E + e];

  __builtin_prefetch(xt + (size_t)col * 64, 0, 0);   // global_prefetch_b8

  float w = 0.f;
  if (sub == 0) {
    const float4 ss = *(const float4*)(s_src + (size_t)row * 4);
    const float4 sd = *(const float4*)(s_dst + (size_t)col * 4);
    const float4 mx = *(const float4*)(seg_max + (size_t)col * 4);
    const float4 dn = *(const float4*)(denom + (size_t)col * 4);
    float sc[4] = { ss.x + sd.x, ss.y + sd.y, ss.z + sd.z, ss.w + sd.w };
    float mm[4] = { mx.x, mx.y, mx.z, mx.w };
    float dd[4] = { dn.x, dn.y, dn.z, dn.w };
#pragma unroll
    for (int h = 0; h < 4; ++h) {
      const float v = sc[h] > 0.f ? sc[h] : ALPHAF * sc[h];
      float m = mm[h];
      if (isinf(m)) m = 0.f;
      const float eh = __expf(v - m);
      w += 0.25f * (eh / (dd[h] + 1e-16f));
    }
  }
  // broadcast from lane 0 / lane 16 of the wave32 to its 16-lane group
  const int src = (threadIdx.x & 31) & ~15;
  w = __shfl(w, src, 32);

  const float4 xv = *(const float4*)(xt + (size_t)col * 64 + sub * 4);
  float* op = out + (size_t)row * 64 + sub * 4;
  atomicAdd(op + 0, w * xv.x);
  atomicAdd(op + 1, w * xv.y);
  atomicAdd(op + 2, w * xv.z);
  atomicAdd(op + 3, w * xv.w);
}

// ---------------------------------------------------------------------------
extern "C" void kernel_launch(void* const* d_in, const int* in_sizes, int n_in,
                              void* d_out, int out_size, void* d_ws, size_t ws_size,
                              hipStream_t stream)
{
  const float* x    = (const float*)d_in[0];          // N*64
  const int*   ei   = (const int*)d_in[1];            // 2*E
  // d_in[2] = edge_weight (unused by reference)
  const float* W    = (const float*)d_in[3];          // 64*64
  const float* avec = (const float*)d_in[4];          // 32

  float* out = (float*)d_out;
  const int N = in_sizes[0] / 64;                     // 100000
  const int E = in_sizes[2];                          // 1600000

  // workspace carve-out (floats): xt | s_src | s_dst | seg_max | denom
  float* ws     = (float*)d_ws;
  float* xt     = ws;
  float* s_src  = xt + (size_t)N * 64;
  float* s_dst  = s_src + (size_t)N * 4;
  float* segm   = s_dst + (size_t)N * 4;
  float* denom  = segm + (size_t)N * 4;

  const int n4 = N * 4, n64 = N * 64;

  ham_wmma_kernel<<<N / 16, 32, 0, stream>>>(x, W, avec, xt, s_src, s_dst);
  init_kernel<<<4096, 256, 0, stream>>>(segm, denom, out, n4, n64);
  segmax_kernel<<<(E + 255) / 256, 256, 0, stream>>>(ei, s_src, s_dst, segm, E);
  denom_kernel<<<(E + 255) / 256, 256, 0, stream>>>(ei, s_src, s_dst, segm, denom, E);
  {
    const long long threads = (long long)E * 16;
    const int blocks = (int)((threads + 255) / 256);
    scatter_kernel<<<blocks, 256, 0, stream>>>(ei, s_src, s_dst, segm, denom, xt, out, E);
  }
}